// MoEGate_1443109011472
// MI455X (gfx1250) — compile-verified
//
#include <hip/hip_runtime.h>
#include <hip/hip_bf16.h>

typedef __attribute__((ext_vector_type(2))) float v2f;
typedef __attribute__((ext_vector_type(8))) float v8f;
typedef __attribute__((ext_vector_type(4))) unsigned int v4u;
typedef __attribute__((ext_vector_type(8))) int v8i;
typedef __attribute__((ext_vector_type(4))) int v4i;

#define NT 16384      // tokens
#define DM 4096       // d_model
#define NE 64         // experts
#define KC 64         // K chunk staged in LDS per TDM transfer
#define WPITCH 68     // floats per W row in LDS (KC + 4 pad via TDM pad fields)
#define WBUF (NE * WPITCH)   // 4352 floats per buffer
#define LPITCH 66     // floats per logits row
#define TOK_PER_WG 128

// Issue one TDM 2D-tile DMA: W[0:64][k0:k0+KC) -> LDS at lds_off, row pitch WPITCH.
// Descriptor packing per CDNA5 ISA (groups 0/1; 2D -> groups 2/3 zero).
__device__ __forceinline__ void tdm_load_w_chunk(const float* Wg, int k0, unsigned lds_off) {
  unsigned long long ga = (unsigned long long)(const void*)Wg + (unsigned long long)k0 * 4ull;
  v4u g0;
  g0[0] = 1u;                                        // count=1, is_restore=0, gather=0
  g0[1] = lds_off;                                   // lds_addr (bytes)
  g0[2] = (unsigned)(ga & 0xFFFFFFFFull);            // global_addr[31:0]
  g0[3] = (unsigned)((ga >> 32) & 0x01FFFFFFull)     // global_addr[56:32]
          | 0x80000000u;                             // type=2 ("image") at bits[127:126]
  v8i g1;
  g1[0] = (2 << 16)      // data_size = 4 bytes
        | (1 << 20)      // pad_enable
        | (5 << 22)      // pad_interval: every 64 DWORDs (one tile row)
        | (3 << 25);     // pad_amount: 4 DWORDs  -> LDS row pitch = 68 floats
  g1[1] = (int)((DM & 0xFFFF) << 16);                // tensor_dim0[15:0] (abar addr = 0)
  g1[2] = (int)(((DM >> 16) & 0xFFFF) | ((NE & 0xFFFF) << 16)); // tdim0[31:16], tdim1[15:0]
  g1[3] = (int)((KC & 0xFFFF) << 16);                // tdim1[31:16]=0, tile_dim0=KC
  g1[4] = NE;                                        // tile_dim1=64, tile_dim2=0
  g1[5] = DM;                                        // tensor_dim0_stride[31:0]
  g1[6] = 0;                                         // stride hi bits / dim1_stride
  g1[7] = 0;
  v4i g2 = {0, 0, 0, 0};
  v4i g3 = {0, 0, 0, 0};
  v8i gx = {0, 0, 0, 0, 0, 0, 0, 0};
  __builtin_amdgcn_tensor_load_to_lds(g0, g1, g2, g3, gx, 0);
}

__global__ __launch_bounds__(256) void moe_gate_gemm(
    const float* __restrict__ x, const float* __restrict__ W,
    float* __restrict__ out_scores, int* __restrict__ out_idx,
    float* __restrict__ expert_acc)
{
  __shared__ float smem[2 * WBUF];   // two TDM W buffers; reused as 128x66 logits tile
  __shared__ float esum[NE];

  const int tid  = threadIdx.x;
  const int wave = tid >> 5;
  const int lane = tid & 31;
  const int tok0 = blockIdx.x * TOK_PER_WG;
  const int wtok = tok0 + wave * 16;

  if (tid < NE) esum[tid] = 0.0f;

  v8f acc0 = {}, acc1 = {}, acc2 = {}, acc3 = {};

  // A fragment: lane L<16 -> row L, K {0,1}; lane L+16 -> row L, K {2,3}
  const int arow  = wtok + (lane & 15);
  const int akoff = (lane >> 4) * 2;
  const float* aptr = x + (size_t)arow * DM + akoff;

  // B fragment (LDS): lane n -> expert col n&15, K off 2*(n>>4)
  const int bcol = lane & 15;
  const int bk   = (lane >> 4) * 2;

  const unsigned lds0 = (unsigned)(size_t)(void*)&smem[0];
  const unsigned lds1 = (unsigned)(size_t)(void*)&smem[WBUF];

  // Prologue: DMA chunk 0 into buffer 0 (TDM ignores EXEC; wave 0 only -> one DMA per WG)
  if (wave == 0) tdm_load_w_chunk(W, 0, lds0);

  int buf = 0;
  for (int k0 = 0; k0 < DM; k0 += KC) {
    if (wave == 0) {
      if (k0 + KC < DM) {  // double-buffer: issue next chunk, wait for current (<=1 outstanding)
        tdm_load_w_chunk(W, k0 + KC, buf ? lds0 : lds1);
        __builtin_amdgcn_s_wait_tensorcnt((short)1);
      } else {
        __builtin_amdgcn_s_wait_tensorcnt((short)0);
      }
    }
    __syncthreads();   // chunk k0 resident in smem[buf]

    __builtin_prefetch(aptr + k0 + KC, 0, 1);   // stream-ahead for x

    const float* wb = &smem[buf * WBUF];
    #pragma unroll 4
    for (int kk = 0; kk < KC; kk += 4) {
      v2f a = *(const v2f*)(aptr + k0 + kk);
      const float* bb = wb + bcol * WPITCH + kk + bk;
      v2f b0 = *(const v2f*)(bb + 0 * 16 * WPITCH);
      v2f b1 = *(const v2f*)(bb + 1 * 16 * WPITCH);
      v2f b2 = *(const v2f*)(bb + 2 * 16 * WPITCH);
      v2f b3 = *(const v2f*)(bb + 3 * 16 * WPITCH);
      acc0 = __builtin_amdgcn_wmma_f32_16x16x4_f32(false, a, false, b0, (short)0, acc0, false, false);
      acc1 = __builtin_amdgcn_wmma_f32_16x16x4_f32(false, a, false, b1, (short)0, acc1, false, false);
      acc2 = __builtin_amdgcn_wmma_f32_16x16x4_f32(false, a, false, b2, (short)0, acc2, false, false);
      acc3 = __builtin_amdgcn_wmma_f32_16x16x4_f32(false, a, false, b3, (short)0, acc3, false, false);
    }
    __syncthreads();   // all waves done with smem[buf^1]'s successor target
    buf ^= 1;
  }

  // Make sure no TDM traffic is still landing before smem is reused for logits
  if (wave == 0) __builtin_amdgcn_s_wait_tensorcnt((short)0);
  __syncthreads();

  // Scatter C fragments -> LDS logits tile (pitch 66: half-waves hit disjoint bank ranges).
  // C layout: VGPR r, lanes 0-15 -> M=r, N=lane; lanes 16-31 -> M=r+8, N=lane-16
  {
    int lrow = wave * 16 + (lane >> 4) * 8;
    int lcol = lane & 15;
    #pragma unroll
    for (int r = 0; r < 8; ++r) {
      smem[(lrow + r) * LPITCH +  0 + lcol] = acc0[r];
      smem[(lrow + r) * LPITCH + 16 + lcol] = acc1[r];
      smem[(lrow + r) * LPITCH + 32 + lcol] = acc2[r];
      smem[(lrow + r) * LPITCH + 48 + lcol] = acc3[r];
    }
  }
  __syncthreads();

  // One thread per token: softmax, top-2 (monotone in logits), renorm, accumulate
  if (tid < TOK_PER_WG) {
    float* lrow = &smem[tid * LPITCH];
    float mx = -1e30f;
    #pragma unroll 8
    for (int e = 0; e < NE; ++e) mx = fmaxf(mx, lrow[e]);

    float sum = 0.0f;
    float m1 = -1e30f, m2 = -1e30f; int i1 = 0, i2 = 0;
    #pragma unroll 8
    for (int e = 0; e < NE; ++e) {
      float l = lrow[e];
      float s = __expf(l - mx);
      lrow[e] = s;                 // cache unnormalized score
      sum += s;
      if (l > m1) { m2 = m1; i2 = i1; m1 = l; i1 = e; }
      else if (l > m2) { m2 = l; i2 = e; }
    }
    float inv = 1.0f / sum;
    #pragma unroll 8
    for (int e = 0; e < NE; ++e) atomicAdd(&esum[e], lrow[e] * inv);

    float e1 = __expf(m1 - mx), e2 = __expf(m2 - mx);
    float rs = 1.0f / (e1 + e2);
    int gt = tok0 + tid;
    out_scores[gt * 2 + 0] = e1 * rs;
    out_scores[gt * 2 + 1] = e2 * rs;
    out_idx[gt * 2 + 0] = i1;
    out_idx[gt * 2 + 1] = i2;
  }
  __syncthreads();

  if (tid < NE) atomicAdd(&expert_acc[tid], esum[tid]);
}

__global__ void moe_gate_loss(const float* __restrict__ expert_acc,
                              float* __restrict__ loss_out)
{
  __shared__ float red[NE];
  int e = threadIdx.x;
  float p = expert_acc[e] * (1.0f / (float)NT);
  red[e] = p * __logf(p + 1e-8f);
  __syncthreads();
  for (int s = 32; s > 0; s >>= 1) {
    if (e < s) red[e] += red[e + s];
    __syncthreads();
  }
  if (e == 0) *loss_out = red[0];
}

__global__ void zero_ws(float* p, int n) {
  int i = blockIdx.x * blockDim.x + threadIdx.x;
  if (i < n) p[i] = 0.0f;
}

extern "C" void kernel_launch(void* const* d_in, const int* in_sizes, int n_in,
                              void* d_out, int out_size, void* d_ws, size_t ws_size,
                              hipStream_t stream) {
  const float* x = (const float*)d_in[0];   // [NT, DM]
  const float* W = (const float*)d_in[1];   // [NE, DM]
  float* out        = (float*)d_out;
  float* out_scores = out;                  // [NT*2]
  int*   out_idx    = (int*)(out + NT * 2); // [NT*2] (int bits)
  float* loss_out   = out + NT * 4;         // [1]
  float* expert_acc = (float*)d_ws;         // [NE]

  zero_ws<<<1, 64, 0, stream>>>(expert_acc, NE);
  moe_gate_gemm<<<NT / TOK_PER_WG, 256, 0, stream>>>(x, W, out_scores, out_idx, expert_acc);
  moe_gate_loss<<<1, NE, 0, stream>>>(expert_acc, loss_out);
}